// SimEncoder_72739566125585
// MI455X (gfx1250) — compile-verified
//
#include <hip/hip_runtime.h>
#include <math.h>

typedef __attribute__((ext_vector_type(2))) float v2f;
typedef __attribute__((ext_vector_type(8))) float v8f;

#define N_NODES 100000
#define E_EDGES 3200000
#define IN_DIM  512
#define EMB     64
#define BATCH   4096

// ---------------------------------------------------------------------------
// fp32 WMMA GEMM: Y[M x Ncols] = X[M x K] * W[Ncols x K]^T + bias
// One wave computes a 16x64 strip (four 16x16 tiles) via V_WMMA_F32_16X16X4_F32,
// so the A fragment is loaded once per k-step and reused by 4 WMMAs.
// A-fragment (ISA 7.12.2, 32-bit A 16x4): lanes 0-15 hold M=0..15,
//   VGPR0 = K {0 | 2}, VGPR1 = K {1 | 3} (split by lane half).
// B is W^T (K x 16), mirrored layout. C/D: VGPR v -> M = v + 8*(lane>=16).
// ---------------------------------------------------------------------------
__global__ __launch_bounds__(32) void wmma_gemm_xwT4(
    const float* __restrict__ X, const float* __restrict__ W,
    const float* __restrict__ bias, float* __restrict__ Y,
    int K, int ldx, int ldw, int ldy) {
  const int lane  = threadIdx.x & 31;
  const size_t row0 = (size_t)blockIdx.x * 16;
  const size_t col0 = (size_t)blockIdx.y * 64;
  const int ml   = lane & 15;
  const int koff = (lane >> 4) << 1;   // lanes 16-31 hold K+2, K+3

  const float* xp  = X + (row0 + ml) * (size_t)ldx + koff;
  const float* wp0 = W + (col0 + ml) * (size_t)ldw + koff;
  const float* wp1 = wp0 + (size_t)16 * ldw;
  const float* wp2 = wp0 + (size_t)32 * ldw;
  const float* wp3 = wp0 + (size_t)48 * ldw;

  v8f c0 = {}, c1 = {}, c2 = {}, c3 = {};
#pragma unroll 2
  for (int k = 0; k < K; k += 4) {
    float2 av = *(const float2*)(xp  + k);
    float2 b0 = *(const float2*)(wp0 + k);
    float2 b1 = *(const float2*)(wp1 + k);
    float2 b2 = *(const float2*)(wp2 + k);
    float2 b3 = *(const float2*)(wp3 + k);
    v2f a; a.x = av.x; a.y = av.y;
    v2f vb;
    // 8 args: (neg_a, A, neg_b, B, c_mod, C, reuse_a, reuse_b)
    vb.x = b0.x; vb.y = b0.y;
    c0 = __builtin_amdgcn_wmma_f32_16x16x4_f32(false, a, false, vb, (short)0, c0, false, false);
    vb.x = b1.x; vb.y = b1.y;
    c1 = __builtin_amdgcn_wmma_f32_16x16x4_f32(false, a, false, vb, (short)0, c1, false, false);
    vb.x = b2.x; vb.y = b2.y;
    c2 = __builtin_amdgcn_wmma_f32_16x16x4_f32(false, a, false, vb, (short)0, c2, false, false);
    vb.x = b3.x; vb.y = b3.y;
    c3 = __builtin_amdgcn_wmma_f32_16x16x4_f32(false, a, false, vb, (short)0, c3, false, false);
  }

  const int mbase = (lane >> 4) * 8;
  float bv0 = 0.0f, bv1 = 0.0f, bv2 = 0.0f, bv3 = 0.0f;
  if (bias) {
    bv0 = bias[col0 + ml];
    bv1 = bias[col0 + 16 + ml];
    bv2 = bias[col0 + 32 + ml];
    bv3 = bias[col0 + 48 + ml];
  }
  float* yp = Y + (row0 + mbase) * (size_t)ldy + col0 + ml;
#pragma unroll
  for (int v = 0; v < 8; ++v) {
    yp[(size_t)v * ldy +  0] = c0[v] + bv0;
    yp[(size_t)v * ldy + 16] = c1[v] + bv1;
    yp[(size_t)v * ldy + 32] = c2[v] + bv2;
    yp[(size_t)v * ldy + 48] = c3[v] + bv3;
  }
}

// ---------------------------------------------------------------------------
// Degree: one thread per edge, float atomic add into deg[row[e]].
// ---------------------------------------------------------------------------
__global__ void deg_kernel(const int* __restrict__ row, float* __restrict__ deg) {
  int e = blockIdx.x * blockDim.x + threadIdx.x;
  if (e < E_EDGES) atomicAdd(&deg[row[e]], 1.0f);
}

// ---------------------------------------------------------------------------
// Pass 1: mu_sum[row] += emb[col].  16 lanes per edge, float4 per lane.
// xdist layout per node: [mu(64) | sigma(64) | gamma(64)]  (ld = 192)
// emb (25.6MB) + xdist (76.8MB) are L2-resident on the 192MB L2, so the
// random gathers and global_atomic_add_f32 scatters resolve in L2.
// ---------------------------------------------------------------------------
__global__ void agg_mu_kernel(const int* __restrict__ row, const int* __restrict__ col,
                              const float* __restrict__ emb, float* __restrict__ xdist) {
  const long long t = (long long)blockIdx.x * blockDim.x + threadIdx.x;
  const int e = (int)(t >> 4);
  const int q = (int)(t & 15);
  if (e >= E_EDGES) return;
  const int rs = row[e], cs = col[e];
  const float4 v = *(const float4*)(emb + (size_t)cs * EMB + q * 4);
  float* dst = xdist + (size_t)rs * (3 * EMB) + q * 4;
  atomicAdd(dst + 0, v.x); atomicAdd(dst + 1, v.y);
  atomicAdd(dst + 2, v.z); atomicAdd(dst + 3, v.w);
}

// mu = where(deg>0, mu_sum/deg, 0)
__global__ void mu_div_kernel(float* __restrict__ xdist, const float* __restrict__ deg) {
  int i = blockIdx.x * blockDim.x + threadIdx.x;  // N*64 threads
  int n = i >> 6, d = i & 63;
  float dg = deg[n];
  float* p = xdist + (size_t)n * (3 * EMB) + d;
  float s = *p;
  *p = (dg > 0.0f) ? (s / dg) : 0.0f;
}

// ---------------------------------------------------------------------------
// Pass 2: sig_sum[row] += (emb[col]-mu[col])^2 ; gam_sum[row] += emb[col]^3
// ---------------------------------------------------------------------------
__global__ void agg_moments_kernel(const int* __restrict__ row, const int* __restrict__ col,
                                   const float* __restrict__ emb, float* __restrict__ xdist) {
  const long long t = (long long)blockIdx.x * blockDim.x + threadIdx.x;
  const int e = (int)(t >> 4);
  const int q = (int)(t & 15);
  if (e >= E_EDGES) return;
  const int rs = row[e], cs = col[e];
  const float4 v = *(const float4*)(emb + (size_t)cs * EMB + q * 4);
  const float4 m = *(const float4*)(xdist + (size_t)cs * (3 * EMB) + q * 4);
  float dx = v.x - m.x, dy = v.y - m.y, dz = v.z - m.z, dw = v.w - m.w;
  float* sig = xdist + (size_t)rs * (3 * EMB) + EMB + q * 4;
  float* gam = xdist + (size_t)rs * (3 * EMB) + 2 * EMB + q * 4;
  atomicAdd(sig + 0, dx * dx); atomicAdd(sig + 1, dy * dy);
  atomicAdd(sig + 2, dz * dz); atomicAdd(sig + 3, dw * dw);
  atomicAdd(gam + 0, v.x * v.x * v.x); atomicAdd(gam + 1, v.y * v.y * v.y);
  atomicAdd(gam + 2, v.z * v.z * v.z); atomicAdd(gam + 3, v.w * v.w * v.w);
}

// sigma = sqrt(where(s==0,1e-16,s)); gamma = cbrt(where(g==0,1e-16,g))
__global__ void moments_fin_kernel(float* __restrict__ xdist, const float* __restrict__ deg) {
  int i = blockIdx.x * blockDim.x + threadIdx.x;  // N*64 threads
  int n = i >> 6, d = i & 63;
  float dg = deg[n];
  float* sp = xdist + (size_t)n * (3 * EMB) + EMB + d;
  float* gp = xdist + (size_t)n * (3 * EMB) + 2 * EMB + d;
  float s = (dg > 0.0f) ? (*sp / dg) : 0.0f;
  s = (s == 0.0f) ? 1e-16f : s;
  *sp = sqrtf(s);
  float g = (dg > 0.0f) ? (*gp / dg) : 0.0f;
  g = (g == 0.0f) ? 1e-16f : g;
  *gp = cbrtf(g);   // == sign(g)*|g|^(1/3)
}

// ---------------------------------------------------------------------------
// Column sums of y[N x 64] for centering (partial per block + atomic fold).
// ---------------------------------------------------------------------------
__global__ void colsum_kernel(const float* __restrict__ Yv, float* __restrict__ colsum, int n) {
  __shared__ float sm[256];
  const int c = threadIdx.x & 63;
  const int g = threadIdx.x >> 6;  // 0..3
  float acc = 0.0f;
  for (int r = blockIdx.x * 4 + g; r < n; r += gridDim.x * 4)
    acc += Yv[(size_t)r * EMB + c];
  sm[threadIdx.x] = acc;
  __syncthreads();
  if (threadIdx.x < 64) {
    float s = sm[threadIdx.x] + sm[threadIdx.x + 64] +
              sm[threadIdx.x + 128] + sm[threadIdx.x + 192];
    atomicAdd(&colsum[threadIdx.x], s);
  }
}

// ---------------------------------------------------------------------------
// Gather batch rows, subtract column mean, L2-normalize (eps=1e-8).
// One wave32 per row: lane handles 2 dims, wave shuffle reduction.
// ---------------------------------------------------------------------------
__global__ void gather_norm_kernel(const float* __restrict__ Yv, const float* __restrict__ colsum,
                                   const int* __restrict__ idx, float* __restrict__ outv,
                                   float invN) {
  const int r = blockIdx.x * 8 + (threadIdx.x >> 5);
  const int lane = threadIdx.x & 31;
  const int n = idx[r];
  float2 v = *(const float2*)(Yv + (size_t)n * EMB + lane * 2);
  v.x -= colsum[lane * 2 + 0] * invN;
  v.y -= colsum[lane * 2 + 1] * invN;
  float ss = v.x * v.x + v.y * v.y;
#pragma unroll
  for (int off = 16; off > 0; off >>= 1)
    ss += __shfl_xor(ss, off, 32);
  float inv = 1.0f / fmaxf(sqrtf(ss), 1e-8f);
  float2 o; o.x = v.x * inv; o.y = v.y * inv;
  *(float2*)(outv + (size_t)r * EMB + lane * 2) = o;
}

// ---------------------------------------------------------------------------
extern "C" void kernel_launch(void* const* d_in, const int* in_sizes, int n_in,
                              void* d_out, int out_size, void* d_ws, size_t ws_size,
                              hipStream_t stream) {
  (void)in_sizes; (void)n_in; (void)out_size; (void)ws_size;
  const float* x         = (const float*)d_in[0];
  const int*   row       = (const int*)d_in[1];
  const int*   col       = (const int*)d_in[2];
  const int*   batch_src = (const int*)d_in[3];
  const int*   batch_tar = (const int*)d_in[4];
  const float* W_pre     = (const float*)d_in[5];
  const float* b_pre     = (const float*)d_in[6];
  const float* W_post    = (const float*)d_in[7];
  const float* b_post    = (const float*)d_in[8];
  float* out = (float*)d_out;

  // Workspace layout (floats): emb | xdist | y | deg | colsum | srcN | tarN
  float* emb    = (float*)d_ws;                         // N*64
  float* xdist  = emb    + (size_t)N_NODES * EMB;       // N*192
  float* y      = xdist  + (size_t)N_NODES * 3 * EMB;   // N*64
  float* deg    = y      + (size_t)N_NODES * EMB;       // N
  float* colsum = deg    + N_NODES;                     // 64
  float* srcN   = colsum + 64;                          // B*64
  float* tarN   = srcN   + (size_t)BATCH * EMB;         // B*64

  // Zero accumulators (graph-capture-safe memset nodes).
  hipMemsetAsync(xdist, 0, (size_t)N_NODES * 3 * EMB * sizeof(float), stream);
  hipMemsetAsync(deg,   0, (size_t)(N_NODES + 64) * sizeof(float), stream);  // deg + colsum

  // 1) emb = x @ W_pre^T + b_pre        [100000 x 64]; x read exactly once
  wmma_gemm_xwT4<<<dim3(N_NODES / 16, EMB / 64), 32, 0, stream>>>(
      x, W_pre, b_pre, emb, IN_DIM, IN_DIM, IN_DIM, EMB);

  // 2) degrees
  deg_kernel<<<(E_EDGES + 255) / 256, 256, 0, stream>>>(row, deg);

  // 3) mu_sum, then mu = mean
  agg_mu_kernel<<<(E_EDGES * 16) / 256, 256, 0, stream>>>(row, col, emb, xdist);
  mu_div_kernel<<<(N_NODES * EMB) / 256, 256, 0, stream>>>(xdist, deg);

  // 4) central 2nd moment + raw 3rd moment, then finalize sigma/gamma
  agg_moments_kernel<<<(E_EDGES * 16) / 256, 256, 0, stream>>>(row, col, emb, xdist);
  moments_fin_kernel<<<(N_NODES * EMB) / 256, 256, 0, stream>>>(xdist, deg);

  // 5) y = [mu|sigma|gamma] @ W_post^T + b_post     [100000 x 64]
  wmma_gemm_xwT4<<<dim3(N_NODES / 16, EMB / 64), 32, 0, stream>>>(
      xdist, W_post, b_post, y, 3 * EMB, 3 * EMB, 3 * EMB, EMB);

  // 6) column sums for centering
  colsum_kernel<<<128, 256, 0, stream>>>(y, colsum, N_NODES);

  // 7) gather + center + normalize batch rows
  gather_norm_kernel<<<BATCH / 8, 256, 0, stream>>>(y, colsum, batch_src, srcN, 1.0f / N_NODES);
  gather_norm_kernel<<<BATCH / 8, 256, 0, stream>>>(y, colsum, batch_tar, tarN, 1.0f / N_NODES);

  // 8) sim = srcN @ tarN^T              [4096 x 4096]
  wmma_gemm_xwT4<<<dim3(BATCH / 16, BATCH / 64), 32, 0, stream>>>(
      srcN, tarN, nullptr, out, EMB, EMB, EMB, BATCH);
}